// TAG_LN_L3_70574902608029
// MI455X (gfx1250) — compile-verified
//
#include <hip/hip_runtime.h>
#include <stdint.h>

typedef float v2f __attribute__((ext_vector_type(2)));
typedef float v8f __attribute__((ext_vector_type(8)));

static constexpr int NN  = 100000;   // nodes
static constexpr int EE  = 1600000;  // edges
static constexpr int HID_ = 128;

// ---------------------------------------------------------------- utilities

__global__ void k_zero(float* __restrict__ p, long long n) {
  long long i = (long long)blockIdx.x * blockDim.x + threadIdx.x;
  if (i < n) p[i] = 0.0f;
}

// deg[col[e]] += ew[e]
__global__ void k_deg(const long long* __restrict__ col,
                      const float* __restrict__ ew,
                      float* __restrict__ deg, int E) {
  int e = blockIdx.x * blockDim.x + threadIdx.x;
  if (e < E) atomicAdd(deg + (int)col[e], ew[e]);
}

// norm[e] = dis[row]*ew*dis[col], dis = deg>0 ? 1/sqrt(deg) : 0
__global__ void k_norm(const long long* __restrict__ row,
                       const long long* __restrict__ col,
                       const float* __restrict__ ew,
                       const float* __restrict__ deg,
                       float* __restrict__ nrm, int E) {
  int e = blockIdx.x * blockDim.x + threadIdx.x;
  if (e >= E) return;
  float dr = deg[(int)row[e]];
  float dc = deg[(int)col[e]];
  float ir = dr > 0.0f ? 1.0f / sqrtf(dr) : 0.0f;
  float ic = dc > 0.0f ? 1.0f / sqrtf(dc) : 0.0f;
  nrm[e] = ir * ew[e] * ic;
}

// dst[col[e],:] += norm[e] * src[row[e],:]   (F = 4<<LOG_CPE floats per row)
// One float4 per thread; for F=128 a whole wave handles one edge -> the full
// 512B row is gathered/scattered coalesced; L2-resident (51MB << 192MB L2).
template <int LOG_CPE>
__global__ void k_spmm(const long long* __restrict__ row,
                       const long long* __restrict__ col,
                       const float* __restrict__ nrm,
                       const float* __restrict__ src,
                       float* __restrict__ dst, int E) {
  constexpr int F = 4 << LOG_CPE;
  long long t = (long long)blockIdx.x * blockDim.x + threadIdx.x;
  int e = (int)(t >> LOG_CPE);
  if (e >= E) return;
  int c = ((int)t & ((1 << LOG_CPE) - 1)) << 2;  // float offset within row
  int r = (int)row[e];
  int d = (int)col[e];
  float w = nrm[e];
  const float4 v = *(const float4*)(src + (long long)r * F + c);
  float* o = dst + (long long)d * F + c;
  atomicAdd(o + 0, w * v.x);
  atomicAdd(o + 1, w * v.y);
  atomicAdd(o + 2, w * v.z);
  atomicAdd(o + 3, w * v.w);
}

// ---------------------------------------------------------------- WMMA GEMM
// Out[M,128] = A[M,K] @ W[K,128] (+ Cin) (+ bias) (ReLU optional)
// Grid: M/16 blocks (M=100000 divides exactly), 256 threads = 8 waves.
// Wave w computes the 16x16 tile at cols [16w,16w+16) via V_WMMA_F32_16X16X4_F32.
// A-frag layout (ISA 7.12.2, 32-bit A 16x4): lane L -> row M=L&15,
//   VGPR0 = K = 2*(L>>4), VGPR1 = K+1  -> one b64 load per k-step.
// C/D layout: VGPR v -> M = v + 8*(L>>4), N = L&15.
__launch_bounds__(256)
__global__ void k_gemm_wmma(const float* __restrict__ A,    // [M,K]
                            const float* __restrict__ W,    // [K,128]
                            const float* __restrict__ Cin,  // [M,128] or null
                            const float* __restrict__ bias, // [128] or null
                            float* __restrict__ Out,        // [M,128]
                            int K, int relu) {
  __shared__ float wlds[128 * 128];  // 64 KB, W row-major (K<=128)

  // cooperative, fully coalesced stage of W into LDS
  for (int idx = threadIdx.x; idx < K * 128; idx += 256) wlds[idx] = W[idx];
  __syncthreads();

  const int lane  = threadIdx.x & 31;
  const int wave  = threadIdx.x >> 5;
  const int m0    = blockIdx.x * 16;
  const int n0    = wave * 16;
  const int ml    = lane & 15;         // A row / B col within tile
  const int khalf = (lane >> 4) << 1;  // 0 or 2

  const float* ap = A + (long long)(m0 + ml) * K + khalf;

  v8f c = {0.f, 0.f, 0.f, 0.f, 0.f, 0.f, 0.f, 0.f};
  const int mrow = m0 + ((lane >> 4) << 3);  // base M for VGPR 0 of C/D
  if (Cin) {
#pragma unroll
    for (int v = 0; v < 8; ++v)
      c[v] = Cin[(long long)(mrow + v) * HID_ + n0 + ml];
  }

  for (int k0 = 0; k0 < K; k0 += 4) {
    v2f a = *(const v2f*)(ap + k0);
    v2f b;
    b.x = wlds[(k0 + khalf) * 128 + n0 + ml];
    b.y = wlds[(k0 + khalf + 1) * 128 + n0 + ml];
    c = __builtin_amdgcn_wmma_f32_16x16x4_f32(
        /*neg_a=*/false, a, /*neg_b=*/false, b,
        /*c_mod=*/(short)0, c, /*reuse_a=*/false, /*reuse_b=*/false);
  }

  const float bv = bias ? bias[n0 + ml] : 0.0f;
#pragma unroll
  for (int v = 0; v < 8; ++v) {
    float val = c[v] + bv;
    if (relu) val = fmaxf(val, 0.0f);
    Out[(long long)(mrow + v) * HID_ + n0 + ml] = val;
  }
}

// out[N,4] = H[N,128] @ Wfc[128,4] + bfc  (memory-bound tail, VALU is fine)
__global__ void k_fc(const float* __restrict__ H,
                     const float* __restrict__ Wfc,
                     const float* __restrict__ bfc,
                     float* __restrict__ out, int n) {
  int i = blockIdx.x * blockDim.x + threadIdx.x;
  if (i >= n) return;
  const float* h = H + (long long)i * HID_;
  float a0 = bfc[0], a1 = bfc[1], a2 = bfc[2], a3 = bfc[3];
#pragma unroll 4
  for (int k = 0; k < HID_; ++k) {
    float hv = h[k];
    a0 += hv * Wfc[k * 4 + 0];
    a1 += hv * Wfc[k * 4 + 1];
    a2 += hv * Wfc[k * 4 + 2];
    a3 += hv * Wfc[k * 4 + 3];
  }
  out[i * 4 + 0] = a0;
  out[i * 4 + 1] = a1;
  out[i * 4 + 2] = a2;
  out[i * 4 + 3] = a3;
}

// ---------------------------------------------------------------- launcher

extern "C" void kernel_launch(void* const* d_in, const int* in_sizes, int n_in,
                              void* d_out, int out_size, void* d_ws, size_t ws_size,
                              hipStream_t stream) {
  const float*     x   = (const float*)d_in[0];
  const long long* ei  = (const long long*)d_in[1];   // int64 [2,E]
  const float*     ea  = (const float*)d_in[2];
  const float*     W0  = (const float*)d_in[3];
  const float*     b0  = (const float*)d_in[4];
  const float*     W1  = (const float*)d_in[5];
  const float*     b1  = (const float*)d_in[6];
  const float*     W2  = (const float*)d_in[7];
  const float*     b2  = (const float*)d_in[8];
  const float*     Wfc = (const float*)d_in[9];
  const float*     bfc = (const float*)d_in[10];
  float*           out = (float*)d_out;

  const long long* row = ei;
  const long long* col = ei + EE;

  // workspace carve-up (~161 MB): deg | norm | ACC | H | P
  float* ws  = (float*)d_ws;
  float* deg = ws;                                // N
  float* nrm = deg + NN;                          // E
  float* ACC = nrm + EE;                          // N*128
  float* H   = ACC + (size_t)NN * HID_;           // N*128
  float* P   = H   + (size_t)NN * HID_;           // N*128

  const int  B       = 256;
  const int  gE      = EE / B;                    // 6250
  const dim3 gGemm(NN / 16);                      // 6250 blocks, 8 waves each
  const long long nNF   = (long long)NN * HID_;
  const long long nN16  = (long long)NN * 16;
  const int  gZN16 = (int)((nN16 + B - 1) / B);
  const int  gZNF  = (int)((nNF  + B - 1) / B);
  const int  gS16  = (EE << 2) / B;               // 25000
  const int  gS128 = (EE << 5) / B;               // 200000

  // ---- gcn_norm
  k_zero<<<(NN + B - 1) / B, B, 0, stream>>>(deg, NN);
  k_deg <<<gE, B, 0, stream>>>(col, ea, deg, EE);
  k_norm<<<gE, B, 0, stream>>>(row, col, ea, deg, nrm, EE);

  // ---- layer 0 (Fin=16, W0 slice stride 16*128), ReLU epilogue -> H
  const int s0 = 16 * HID_;
  k_gemm_wmma<<<gGemm, B, 0, stream>>>(x, W0 + 0 * s0, nullptr, nullptr, ACC, 16, 0);
  k_zero<<<gZN16, B, 0, stream>>>(P, nN16);
  k_spmm<2><<<gS16, B, 0, stream>>>(row, col, nrm, x, P, EE);
  k_gemm_wmma<<<gGemm, B, 0, stream>>>(P, W0 + 1 * s0, ACC, nullptr, ACC, 16, 0);
  k_zero<<<gZN16, B, 0, stream>>>(H, nN16);
  k_spmm<2><<<gS16, B, 0, stream>>>(row, col, nrm, P, H, EE);
  k_gemm_wmma<<<gGemm, B, 0, stream>>>(H, W0 + 2 * s0, ACC, nullptr, ACC, 16, 0);
  k_zero<<<gZN16, B, 0, stream>>>(P, nN16);
  k_spmm<2><<<gS16, B, 0, stream>>>(row, col, nrm, H, P, EE);
  k_gemm_wmma<<<gGemm, B, 0, stream>>>(P, W0 + 3 * s0, ACC, b0, H, 16, 1);

  // ---- layers 1 & 2 (Fin=128, slice stride 128*128); layer2 has no ReLU
  const int s1 = HID_ * HID_;
  const float* Ws[2] = {W1, W2};
  const float* bs[2] = {b1, b2};
  for (int l = 0; l < 2; ++l) {
    const float* W = Ws[l];
    const int    rl = (l == 0) ? 1 : 0;
    k_gemm_wmma<<<gGemm, B, 0, stream>>>(H, W + 0 * s1, nullptr, nullptr, ACC, HID_, 0);
    k_zero<<<gZNF, B, 0, stream>>>(P, nNF);
    k_spmm<5><<<gS128, B, 0, stream>>>(row, col, nrm, H, P, EE);
    k_gemm_wmma<<<gGemm, B, 0, stream>>>(P, W + 1 * s1, ACC, nullptr, ACC, HID_, 0);
    k_zero<<<gZNF, B, 0, stream>>>(H, nNF);
    k_spmm<5><<<gS128, B, 0, stream>>>(row, col, nrm, P, H, EE);
    k_gemm_wmma<<<gGemm, B, 0, stream>>>(H, W + 2 * s1, ACC, nullptr, ACC, HID_, 0);
    k_zero<<<gZNF, B, 0, stream>>>(P, nNF);
    k_spmm<5><<<gS128, B, 0, stream>>>(row, col, nrm, H, P, EE);
    k_gemm_wmma<<<gGemm, B, 0, stream>>>(P, W + 3 * s1, ACC, bs[l], H, HID_, rl);
  }

  // ---- final FC
  k_fc<<<(NN + B - 1) / B, B, 0, stream>>>(H, Wfc, bfc, out, NN);
}